// DepthNet_9096740733671
// MI455X (gfx1250) — compile-verified
//
#include <hip/hip_runtime.h>
#include <hip/hip_bf16.h>
#include <stdint.h>

#define Bv 1
#define Vv 5
#define Cc 32
#define Dd 48
#define Hh 128
#define Ww 160
#define HW (Hh*Ww)          // 20480
#define DHW (Dd*HW)         // 983040

typedef float v2f __attribute__((ext_vector_type(2)));
typedef float v8f __attribute__((ext_vector_type(8)));

__device__ __forceinline__ int iminD(int a, int b){ return a < b ? a : b; }
__device__ __forceinline__ int imaxD(int a, int b){ return a > b ? a : b; }

// ---------------------------------------------------------------------------
// Kernel A: build per-view [rot(9) | trans(3)] from proj matrices (1 thread).
// proj_new[v] = [K[:3,:3] @ E[:3,:4] ; E[3,:]];  proj = proj_new[v] @ inv(proj_new[0])
// inv of [M t; 0 1] = [M^-1, -M^-1 t; 0 1]
// ---------------------------------------------------------------------------
__global__ void dn_setup_rt(const float* __restrict__ pm, float* __restrict__ rt)
{
    if (threadIdx.x != 0 || blockIdx.x != 0) return;
    float P[Vv][16];
    for (int v = 0; v < Vv; ++v) {
        const float* E = pm + (v*2 + 0)*16;
        const float* K = pm + (v*2 + 1)*16;
        for (int i = 0; i < 3; ++i)
            for (int j = 0; j < 4; ++j) {
                float s = 0.f;
                for (int k = 0; k < 3; ++k) s += K[i*4+k] * E[k*4+j];
                P[v][i*4+j] = s;
            }
        for (int j = 0; j < 4; ++j) P[v][12+j] = E[12+j];
    }
    const float* R = P[0];
    float m00=R[0], m01=R[1], m02=R[2],  t0=R[3];
    float m10=R[4], m11=R[5], m12=R[6],  t1=R[7];
    float m20=R[8], m21=R[9], m22=R[10], t2=R[11];
    float det = m00*(m11*m22 - m12*m21) - m01*(m10*m22 - m12*m20) + m02*(m10*m21 - m11*m20);
    float id = 1.0f / det;
    float i00 =  (m11*m22 - m12*m21)*id;
    float i01 = -(m01*m22 - m02*m21)*id;
    float i02 =  (m01*m12 - m02*m11)*id;
    float i10 = -(m10*m22 - m12*m20)*id;
    float i11 =  (m00*m22 - m02*m20)*id;
    float i12 = -(m00*m12 - m02*m10)*id;
    float i20 =  (m10*m21 - m11*m20)*id;
    float i21 = -(m00*m21 - m01*m20)*id;
    float i22 =  (m00*m11 - m01*m10)*id;
    float v0 = -(i00*t0 + i01*t1 + i02*t2);
    float v1 = -(i10*t0 + i11*t1 + i12*t2);
    float v2 = -(i20*t0 + i21*t1 + i22*t2);
    float IR[16] = { i00,i01,i02,v0,  i10,i11,i12,v1,  i20,i21,i22,v2,  0.f,0.f,0.f,1.f };
    for (int v = 1; v < Vv; ++v) {
        float Pr[16];
        for (int i = 0; i < 4; ++i)
            for (int j = 0; j < 4; ++j) {
                float s = 0.f;
                for (int k = 0; k < 4; ++k) s += P[v][i*4+k] * IR[k*4+j];
                Pr[i*4+j] = s;
            }
        float* o = rt + (v-1)*12;
        o[0]=Pr[0]; o[1]=Pr[1]; o[2]=Pr[2];
        o[3]=Pr[4]; o[4]=Pr[5]; o[5]=Pr[6];
        o[6]=Pr[8]; o[7]=Pr[9]; o[8]=Pr[10];
        o[9]=Pr[3]; o[10]=Pr[7]; o[11]=Pr[11];
    }
}

// ---------------------------------------------------------------------------
// Kernel B: fused warp + bilinear gather + adaptive aggregation.
// One thread per (d, y, x) sample; loops views and the 32 channels in regs.
// Constants (rot/trans + w1) staged into LDS via CDNA5 async-to-LDS loads.
// Writes cost_agg [C, DHW].
// ---------------------------------------------------------------------------
__global__ __launch_bounds__(256) void dn_cost_volume(
    const float* __restrict__ feat, const float* __restrict__ dvals,
    const float* __restrict__ rt,   const float* __restrict__ w1,
    const float* __restrict__ b1,   const float* __restrict__ w2,
    const float* __restrict__ b2,   float* __restrict__ cost_out)
{
    __shared__ float s_rt[48];
    __shared__ float s_w1[Cc];

    unsigned t = threadIdx.x;
    if (t < 48u + Cc) {
        unsigned lds;
        unsigned long long ga;
        if (t < 48u) {
            lds = (unsigned)(uintptr_t)(&s_rt[t]);
            ga  = (unsigned long long)(uintptr_t)(rt + t);
        } else {
            lds = (unsigned)(uintptr_t)(&s_w1[t - 48u]);
            ga  = (unsigned long long)(uintptr_t)(w1 + (t - 48u));
        }
        // CDNA5 async global->LDS (ASYNCcnt path)
        asm volatile("global_load_async_to_lds_b32 %0, %1, off"
                     :: "v"(lds), "v"(ga) : "memory");
    }
    asm volatile("s_wait_asynccnt 0" ::: "memory");
    __syncthreads();

    int s   = blockIdx.x * blockDim.x + threadIdx.x;   // grid exactly covers DHW
    int d   = s / HW;
    int pix = s - d * HW;
    int y   = pix / Ww;
    int x   = pix - y * Ww;

    float ref[Cc];
#pragma unroll
    for (int c = 0; c < Cc; ++c) ref[c] = feat[c*HW + pix];   // view 0

    float va[Cc];
#pragma unroll
    for (int c = 0; c < Cc; ++c) va[c] = 0.f;

    const float dv  = dvals[(size_t)d*HW + pix];
    const float b1v = b1[0], w2v = w2[0], b2v = b2[0];
    const float fx = (float)x, fy = (float)y;

    for (int v = 1; v < Vv; ++v) {
        const float* r = &s_rt[(v-1)*12];
        float X = (r[0]*fx + r[1]*fy + r[2]) * dv + r[9];
        float Y = (r[3]*fx + r[4]*fy + r[5]) * dv + r[10];
        float Z = (r[6]*fx + r[7]*fy + r[8]) * dv + r[11];
        float zz = (fabsf(Z) < 1e-6f) ? 1e-6f : Z;
        float px = X / zz, py = Y / zz;
        float x0f = floorf(px), y0f = floorf(py);
        float wx = px - x0f, wy = py - y0f;
        int xi = (int)x0f, yi = (int)y0f;

        float inx0 = (xi   >= 0 && xi   <= Ww-1) ? 1.f : 0.f;
        float inx1 = (xi+1 >= 0 && xi+1 <= Ww-1) ? 1.f : 0.f;
        float iny0 = (yi   >= 0 && yi   <= Hh-1) ? 1.f : 0.f;
        float iny1 = (yi+1 >= 0 && yi+1 <= Hh-1) ? 1.f : 0.f;

        int xc0 = iminD(imaxD(xi,   0), Ww-1);
        int xc1 = iminD(imaxD(xi+1, 0), Ww-1);
        int yc0 = iminD(imaxD(yi,   0), Hh-1);
        int yc1 = iminD(imaxD(yi+1, 0), Hh-1);
        int i00 = yc0*Ww + xc0, i01 = yc0*Ww + xc1;
        int i10 = yc1*Ww + xc0, i11 = yc1*Ww + xc1;

        float ww00 = (1.f-wx)*(1.f-wy) * (inx0*iny0);
        float ww01 =        wx*(1.f-wy) * (inx1*iny0);
        float ww10 = (1.f-wx)*       wy * (inx0*iny1);
        float ww11 =        wx*       wy * (inx1*iny1);

        const float* sf = feat + (size_t)v*Cc*HW;
        float vol[Cc];
        float dot = 0.f;
#pragma unroll
        for (int c = 0; c < Cc; ++c) {
            const float* fc = sf + (size_t)c*HW;
            float wp = fc[i00]*ww00 + fc[i01]*ww01 + fc[i10]*ww10 + fc[i11]*ww11;
            float df = ref[c] - wp;
            float q  = df * df;
            vol[c] = q;
            dot = fmaf(q, s_w1[c], dot);
        }
        float h   = fmaxf(dot + b1v, 0.f);
        float wgt = fmaxf(fmaf(h, w2v, b2v), 0.f) + 1.f;
#pragma unroll
        for (int c = 0; c < Cc; ++c) va[c] = fmaf(wgt, vol[c], va[c]);
    }

    const float inv_nv = 1.0f / (float)(Vv - 1);
#pragma unroll
    for (int c = 0; c < Cc; ++c)
        cost_out[(size_t)c*DHW + s] = va[c] * inv_nv;
}

// ---------------------------------------------------------------------------
// Kernel C: cost_reg[s] = sum_c cost_agg[c,s]*wr[c] + br  via V_WMMA_F32_16X16X4_F32.
// Each wave handles 16 samples; 8 chunked WMMAs accumulate the 32-channel dot.
// A (16x4 f32): lanes 0-15 hold K=0,1; lanes 16-31 hold K=2,3 (same 16 rows).
// B (4x16): wr chunk replicated across all 16 columns.
// ---------------------------------------------------------------------------
__global__ __launch_bounds__(256) void dn_costreg_wmma(
    const float* __restrict__ cost, const float* __restrict__ wr,
    const float* __restrict__ br,   float* __restrict__ creg)
{
    int tid  = blockIdx.x * blockDim.x + threadIdx.x;
    int wave = tid >> 5;
    int lane = threadIdx.x & 31;
    int s0   = wave << 4;          // 16 samples per wave; grid covers DHW exactly
    int half = lane >> 4;
    int p    = lane & 15;

    v8f acc = {0.f,0.f,0.f,0.f,0.f,0.f,0.f,0.f};
#pragma unroll
    for (int k = 0; k < Cc; k += 4) {
        int ca = k + 2*half;
        v2f a, b;
        a.x = cost[(size_t)(ca+0)*DHW + s0 + p];
        a.y = cost[(size_t)(ca+1)*DHW + s0 + p];
        b.x = wr[ca + 0];
        b.y = wr[ca + 1];
        acc = __builtin_amdgcn_wmma_f32_16x16x4_f32(
            /*neg_a=*/false, a, /*neg_b=*/false, b,
            /*c_mod=*/(short)0, acc, /*reuse_a=*/false, /*reuse_b=*/false);
    }

    // All 16 columns of D are identical; extract row m = lane's own column.
    // lanes 0-7 -> rows 0-7 (VGPR i = row i); lanes 16-23 -> rows 8-15.
    int q = lane & 7;
    float r = (q < 4)
        ? ((q < 2) ? ((q == 0) ? acc[0] : acc[1]) : ((q == 2) ? acc[2] : acc[3]))
        : ((q < 6) ? ((q == 4) ? acc[4] : acc[5]) : ((q == 6) ? acc[6] : acc[7]));
    if ((lane & 8) == 0) {
        int m = (half << 3) | q;
        creg[s0 + m] = r + br[0];
    }
}

// ---------------------------------------------------------------------------
// Kernel D: per-pixel softmax over D, prob_volume, depth expectation, confidence.
// cost_reg lives (temporarily) in the prob_volume output slot; overwritten in place.
// ---------------------------------------------------------------------------
__global__ __launch_bounds__(256) void dn_softmax_depth(
    const float* __restrict__ dvals, float* __restrict__ out)
{
    int pix = blockIdx.x * blockDim.x + threadIdx.x;   // grid covers HW exactly
    float* depth = out;
    float* conf  = out + HW;
    float* prob  = out + 2*HW;

    float cr[Dd];
    float mx = -3.4e38f;
#pragma unroll
    for (int d = 0; d < Dd; ++d) {
        cr[d] = prob[(size_t)d*HW + pix];
        mx = fmaxf(mx, cr[d]);
    }
    float ssum = 0.f;
#pragma unroll
    for (int d = 0; d < Dd; ++d) {
        float e = expf(cr[d] - mx);
        cr[d] = e;
        ssum += e;
    }
    float inv = 1.0f / ssum;
    float de = 0.f, pmax = 0.f;
#pragma unroll
    for (int d = 0; d < Dd; ++d) {
        float pr = cr[d] * inv;
        prob[(size_t)d*HW + pix] = pr;
        de = fmaf(pr, dvals[(size_t)d*HW + pix], de);
        pmax = fmaxf(pmax, pr);
    }
    depth[pix] = de;
    conf[pix]  = pmax;
}

// ---------------------------------------------------------------------------
extern "C" void kernel_launch(void* const* d_in, const int* in_sizes, int n_in,
                              void* d_out, int out_size, void* d_ws, size_t ws_size,
                              hipStream_t stream)
{
    const float* feat  = (const float*)d_in[0];
    const float* pm    = (const float*)d_in[1];
    const float* dvals = (const float*)d_in[2];
    const float* w1    = (const float*)d_in[3];
    const float* b1    = (const float*)d_in[4];
    const float* w2    = (const float*)d_in[5];
    const float* b2    = (const float*)d_in[6];
    const float* wr    = (const float*)d_in[7];
    const float* br    = (const float*)d_in[8];
    float* out = (float*)d_out;
    float* rt  = (float*)d_ws;                   // 48 floats of scratch

    float* cost = out + 2*HW + DHW;              // cost_agg output slot
    float* creg = out + 2*HW;                    // prob_volume slot (holds cost_reg first)

    dn_setup_rt<<<1, 1, 0, stream>>>(pm, rt);
    dn_cost_volume<<<DHW/256, 256, 0, stream>>>(feat, dvals, rt, w1, b1, w2, b2, cost);
    dn_costreg_wmma<<<(DHW/16)/8, 256, 0, stream>>>(cost, wr, br, creg);  // 8 waves/block
    dn_softmax_depth<<<HW/256, 256, 0, stream>>>(dvals, out);
}